// ContinuousTimeModel_50208167690605
// MI455X (gfx1250) — compile-verified
//
#include <hip/hip_runtime.h>

typedef float v2f __attribute__((ext_vector_type(2)));
typedef float v8f __attribute__((ext_vector_type(8)));

#define TT  512
#define BB  128
#define DIN 512
#define HH  1024
static constexpr float CCONST = 1.0f / 10.0f;   // DT/TAU

// C(MxN) = A(MxK) * W(NxK)^T (+ epilogue), all f32 row-major.
// Exact-precision path via V_WMMA_F32_16X16X4_F32.
// Block tile 128x128, BK=32, 256 threads = 8 waves (2 in M x 4 in N),
// each wave computes a 64x32 subtile (4x2 fragments of 16x16).
// Double-buffered LDS: next tile's global loads are issued before the
// compute phase, ds_store lands in the alternate buffer, 1 barrier/tile.
// MODE 0: out = acc + bias                                  (hterm kernel)
// MODE 1: out = hidden*(1-c) + c*relu(acc + bias + hterm)   (main kernel)
template <int MODE>
__global__ __launch_bounds__(256) void gemm_nt_wmma_f32(
    const float* __restrict__ A, const float* __restrict__ W, int M, int K,
    const float* __restrict__ bias,    // [HH]
    const float* __restrict__ hidden,  // [BB,HH]  (MODE 1)
    const float* __restrict__ hterm,   // [BB,HH]  (MODE 1)
    float* __restrict__ out)           // [M,HH]
{
  constexpr int BM = 128, BN = 128, BK = 32;
  constexpr int LP = BK + 2;  // pitch 34 dwords -> conflict-free frag reads
  __shared__ float lA[2][BM * LP];
  __shared__ float lW[2][BN * LP];

  const int tid   = threadIdx.x;
  const int wave  = tid >> 5;
  const int lane  = tid & 31;
  const int l16   = lane & 15;
  const int kh    = lane >> 4;             // selects K pair within 4-wide frag
  const int mBase = (int)blockIdx.x * BM;
  const int nBase = (int)blockIdx.y * BN;
  const int aOff  = (wave & 1) * 64;       // wave M offset within block tile
  const int bOff  = (wave >> 1) * 32;      // wave N offset within block tile

  v8f acc[4][2];
#pragma unroll
  for (int i = 0; i < 4; ++i)
#pragma unroll
    for (int j = 0; j < 2; ++j) acc[i][j] = (v8f){};

  // Issue this tile's global loads (kept in VGPRs until the ds_store phase).
  auto gload = [&](int kb, float4 va[4], float4 vw[4]) {
#pragma unroll
    for (int it = 0; it < 4; ++it) {
      const int idx = it * 256 + tid;      // 0..1023
      const int r   = idx >> 3;            // row 0..127
      const int c   = (idx & 7) << 2;      // col 0..28 step 4
      va[it] = *(const float4*)&A[(size_t)(mBase + r) * K + kb + c];
      vw[it] = *(const float4*)&W[(size_t)(nBase + r) * K + kb + c];
    }
  };

  auto lstore = [&](int buf, const float4 va[4], const float4 vw[4]) {
#pragma unroll
    for (int it = 0; it < 4; ++it) {
      const int idx = it * 256 + tid;
      const int r   = idx >> 3;
      const int c   = (idx & 7) << 2;
      float* da = &lA[buf][r * LP + c];
      *(float2*)(da)     = make_float2(va[it].x, va[it].y);
      *(float2*)(da + 2) = make_float2(va[it].z, va[it].w);
      float* dw = &lW[buf][r * LP + c];
      *(float2*)(dw)     = make_float2(vw[it].x, vw[it].y);
      *(float2*)(dw + 2) = make_float2(vw[it].z, vw[it].w);
    }
  };

  auto compute = [&](int buf) {
#pragma unroll
    for (int kk = 0; kk < BK / 4; ++kk) {
      const int k0 = kk * 4 + kh * 2;
      v2f a[4], b[2];
#pragma unroll
      for (int i = 0; i < 4; ++i)
        a[i] = *(const v2f*)&lA[buf][(aOff + i * 16 + l16) * LP + k0];
#pragma unroll
      for (int j = 0; j < 2; ++j)
        b[j] = *(const v2f*)&lW[buf][(bOff + j * 16 + l16) * LP + k0];
#pragma unroll
      for (int i = 0; i < 4; ++i)
#pragma unroll
        for (int j = 0; j < 2; ++j)
          acc[i][j] = __builtin_amdgcn_wmma_f32_16x16x4_f32(
              false, a[i], false, b[j], (short)0, acc[i][j], false, false);
    }
  };

  const int nt = K / BK;
  float4 va[4], vw[4];

  gload(0, va, vw);
  lstore(0, va, vw);
  __syncthreads();

  for (int t = 1; t < nt; ++t) {
    gload(t * BK, va, vw);               // overlap global latency w/ compute
    if (t + 1 < nt) {                    // speculative prefetch, 1 tile ahead
      const int kp = (t + 1) * BK;
#pragma unroll
      for (int it = 0; it < 4; ++it) {
        const int idx = it * 256 + tid;
        const int r   = idx >> 3;
        const int c   = (idx & 7) << 2;
        __builtin_prefetch(&A[(size_t)(mBase + r) * K + kp + c], 0, 3);
        __builtin_prefetch(&W[(size_t)(nBase + r) * K + kp + c], 0, 3);
      }
    }
    compute((t - 1) & 1);
    lstore(t & 1, va, vw);               // other buffer: no RAW w/ compute
    __syncthreads();                     // publish stores; fence prev buffer
  }
  compute((nt - 1) & 1);

  // Epilogue. D layout: VGPR r, lane l -> M = r + 8*(l>>4), N = l&15.
#pragma unroll
  for (int i = 0; i < 4; ++i) {
#pragma unroll
    for (int j = 0; j < 2; ++j) {
      const int col = nBase + bOff + j * 16 + l16;
#pragma unroll
      for (int r = 0; r < 8; ++r) {
        const int row = mBase + aOff + i * 16 + kh * 8 + r;
        const float v = acc[i][j][r];
        if (MODE == 0) {
          out[(size_t)row * HH + col] = v + bias[col];
        } else {
          const int    bidx = row & (BB - 1);  // row % B
          const size_t hIdx = (size_t)bidx * HH + col;
          const float  pre  = v + bias[col] + hterm[hIdx];
          const float  o    = hidden[hIdx] * (1.0f - CCONST) +
                              CCONST * fmaxf(pre, 0.0f);
          out[(size_t)row * HH + col] = o;
        }
      }
    }
  }
}

extern "C" void kernel_launch(void* const* d_in, const int* in_sizes, int n_in,
                              void* d_out, int out_size, void* d_ws,
                              size_t ws_size, hipStream_t stream) {
  const float* x      = (const float*)d_in[0];  // [T,B,DIN]
  const float* hidden = (const float*)d_in[1];  // [B,H]
  const float* W_in   = (const float*)d_in[2];  // [H,DIN]
  const float* b_in   = (const float*)d_in[3];  // [H]
  const float* W_h    = (const float*)d_in[4];  // [H,H]
  const float* b_h    = (const float*)d_in[5];  // [H]
  float*       out    = (float*)d_out;          // [T*B*H] ++ [B*H]
  float*       hterm  = (float*)d_ws;           // [B,H] scratch (512 KB)

  // 1) hterm = hidden @ W_h^T + b_h      (M=128, N=1024, K=1024)
  dim3 g0(BB / 128, HH / 128);
  gemm_nt_wmma_f32<0><<<g0, 256, 0, stream>>>(hidden, W_h, BB, HH, b_h,
                                              nullptr, nullptr, hterm);

  // 2) out = hidden*(1-c) + c*relu(x @ W_in^T + b_in + hterm)
  //    (M = T*B = 65536, N=1024, K=512)
  dim3 g1((TT * BB) / 128, HH / 128);
  gemm_nt_wmma_f32<1><<<g1, 256, 0, stream>>>(x, W_in, TT * BB, DIN, b_in,
                                              hidden, hterm, out);

  // 3) Second tuple element: hidden passthrough appended to d_out.
  hipMemcpyAsync(out + (size_t)TT * BB * HH, hidden,
                 (size_t)BB * HH * sizeof(float), hipMemcpyDeviceToDevice,
                 stream);
}